// NasAutoGraphC_36816459661705
// MI455X (gfx1250) — compile-verified
//
#include <hip/hip_runtime.h>
#include <math.h>

// CDNA5 (gfx1250) wave32 WMMA f32 path for a 2-cell Cheb+SAGE GNN.
// - All GEMMs use V_WMMA_F32_16X16X4_F32 (exact fp32 matrix pipe).
// - Weights pre-packed into WMMA B-fragment lane order -> coalesced b64 loads.
// - Edge aggregation: one wave per edge, scalar (SMEM) index loads,
//   float4 feature moves + global f32 atomics (L2-resident buffers).

typedef __attribute__((ext_vector_type(2))) float v2f;
typedef __attribute__((ext_vector_type(8))) float v8f;

#define LRELU_SLOPE 0.01f

// ---------------- elementwise / graph-prep kernels ----------------

__global__ __launch_bounds__(256) void k_fill(float* __restrict__ p, float v, int n) {
  int i = blockIdx.x * 256 + threadIdx.x;
  if (i < n) p[i] = v;
}

__global__ __launch_bounds__(256) void k_edge_prep(const int* __restrict__ src,
                                                   const int* __restrict__ dst,
                                                   const float* __restrict__ w,
                                                   float* __restrict__ deg,
                                                   float* __restrict__ cnt,
                                                   float* __restrict__ loopw, int E) {
  int e = blockIdx.x * 256 + threadIdx.x;
  if (e >= E) return;
  int s = src[e], d = dst[e];
  float wv = w[e];
  if (s == d) {
    loopw[s] = wv;                 // SAGE keeps existing self-loop weight
  } else {
    atomicAdd(&deg[s], wv);        // Cheb: deg = segment_sum(wc, src)
    atomicAdd(&cnt[d], 1.0f);      // SAGE: non-loop in-degree per dst
  }
}

__global__ __launch_bounds__(256) void k_node_finalize(float* __restrict__ deg_dis,
                                                       float* __restrict__ cnt_inv, int n) {
  int i = blockIdx.x * 256 + threadIdx.x;
  if (i >= n) return;
  float dg = deg_dis[i];
  deg_dis[i] = (dg > 0.f) ? rsqrtf(dg) : 0.f;   // dis
  cnt_inv[i] = 1.0f / (cnt_inv[i] + 1.0f);      // 1/(cnt+1)
}

__global__ __launch_bounds__(256) void k_edge_norm(const int* __restrict__ src,
                                                   const int* __restrict__ dst,
                                                   const float* __restrict__ w,
                                                   const float* __restrict__ dis,
                                                   float* __restrict__ normc,
                                                   float* __restrict__ wec, int E) {
  int e = blockIdx.x * 256 + threadIdx.x;
  if (e >= E) return;
  int s = src[e], d = dst[e];
  float wv = w[e];
  float wc = (s == d) ? 0.f : wv;
  normc[e] = -dis[s] * wc * dis[d];   // L_hat off-diagonal
  wec[e] = wc;                        // SAGE non-loop weights
}

// OUT[dst[e], :] += coeff[e] * X[src[e], :]   (H == 128)
// One wave per edge: e is wave-uniform -> scalar loads for src/dst/coeff;
// 32 lanes x float4 = 128 features.
__global__ __launch_bounds__(256) void k_scatter(const int* __restrict__ src,
                                                 const int* __restrict__ dst,
                                                 const float* __restrict__ coeff,
                                                 const float* __restrict__ X,
                                                 float* __restrict__ OUT, int E) {
  int e = blockIdx.x * 8 + (int)(threadIdx.x >> 5);
  if (e >= E) return;
  e = __builtin_amdgcn_readfirstlane(e);
  const int lane = threadIdx.x & 31;
  const float c = coeff[e];
  if (c == 0.f) return;              // self-loops removed (uniform branch)
  const int s = src[e];
  const int d = dst[e];
  const float4 v = *((const float4*)(X + (size_t)s * 128) + lane);
  float* o = OUT + (size_t)d * 128 + lane * 4;
  atomicAdd(o + 0, c * v.x);
  atomicAdd(o + 1, c * v.y);
  atomicAdd(o + 2, c * v.z);
  atomicAdd(o + 3, c * v.w);
}

// S = (S + loopw[m] * XH) * invcnt[m]   (float4 over H == 128)
__global__ __launch_bounds__(256) void k_sage_combine(float* __restrict__ S,
                                                      const float* __restrict__ XH,
                                                      const float* __restrict__ loopw,
                                                      const float* __restrict__ invc, int n) {
  unsigned gid = blockIdx.x * 256u + threadIdx.x;
  int m = (int)(gid >> 5), l = (int)(gid & 31u);
  if (m >= n) return;
  size_t idx = (size_t)m * 128 + (size_t)l * 4;
  float lw = loopw[m], ic = invc[m];
  float4 s = *(float4*)(S + idx);
  float4 xh = *(const float4*)(XH + idx);
  s.x = (s.x + lw * xh.x) * ic;
  s.y = (s.y + lw * xh.y) * ic;
  s.z = (s.z + lw * xh.z) * ic;
  s.w = (s.w + lw * xh.w) * ic;
  *(float4*)(S + idx) = s;
}

__global__ __launch_bounds__(256) void k_logsoftmax(const float* __restrict__ L,
                                                    float* __restrict__ out, int n, int C) {
  int i = blockIdx.x * 256 + threadIdx.x;
  if (i >= n) return;
  const float* r = L + (size_t)i * C;
  float mx = r[0];
  for (int c = 1; c < C; ++c) mx = fmaxf(mx, r[c]);
  float s = 0.f;
  for (int c = 0; c < C; ++c) s += expf(r[c] - mx);
  float lse = mx + logf(s);
  float* o = out + (size_t)i * C;
  for (int c = 0; c < C; ++c) o[c] = r[c] - lse;
}

// ---------------- weight packing into WMMA B-fragment order ----------------
// V_WMMA_F32_16X16X4_F32 B layout: lane L, reg j -> B[K = k + 2*(L/16) + j][N = L%16].
// Packed: Wp[((k4*NT + t)*32 + lane)] = v2f{ W[kk][n], W[kk+1][n] },
//   kk = 4*k4 + 2*(lane/16), n = t*16 + lane%16  (zeros past NDIM).
__global__ __launch_bounds__(256) void k_pack_w(const float* __restrict__ W,
                                                float* __restrict__ Wp,
                                                int K, int NDIM, int NT) {
  int i = blockIdx.x * 256 + threadIdx.x;   // one thread per v2f fragment
  int total = (K >> 2) * NT * 32;
  if (i >= total) return;
  int lane = i & 31;
  int t = (i >> 5) % NT;
  int k4 = (i >> 5) / NT;
  int n = t * 16 + (lane & 15);
  int kk = k4 * 4 + 2 * (lane >> 4);
  float b0 = 0.f, b1 = 0.f;
  if (n < NDIM) {
    b0 = W[(size_t)kk * NDIM + n];
    b1 = W[(size_t)(kk + 1) * NDIM + n];
  }
  Wp[(size_t)i * 2 + 0] = b0;
  Wp[(size_t)i * 2 + 1] = b1;
}

// ---------------- WMMA f32 GEMM ----------------
// D[M x NDIM] = epilogue( A1[M x K] @ W1 (+ A2 @ W2) + bias ), W pre-packed.
// One wave per 16-row M-strip; NT 16-col tiles of v8f accumulators in registers.
//   A: lane L, reg j -> A[M = L%16][K = k + 2*(L/16) + j]   (contiguous v2f)
//   D: lane L, reg v -> D[M = v + 8*(L/16)][N = L%16]
template <int NT, int NDIM, bool HAS_A2, bool LEAKY, bool ACCUM>
__global__ __launch_bounds__(256) void k_gemm_wmma(
    const float* __restrict__ A1, const v2f* __restrict__ Wp1,
    const float* __restrict__ A2, const v2f* __restrict__ Wp2,
    const float* __restrict__ bias, float* __restrict__ D, int M, int K) {
  const int lane = threadIdx.x & 31;
  const int wave = threadIdx.x >> 5;
  const int mtile = blockIdx.x * 8 + wave;
  if (mtile * 16 >= M) return;          // uniform per-wave exit: EXEC stays all-ones
  const int lrow = lane & 15;
  const int khalf = lane >> 4;

  const v8f vzero = {0.f, 0.f, 0.f, 0.f, 0.f, 0.f, 0.f, 0.f};
  v8f acc[NT];
#pragma unroll
  for (int t = 0; t < NT; ++t) acc[t] = vzero;

  for (int pass = 0; pass < (HAS_A2 ? 2 : 1); ++pass) {
    const float* A = (pass == 0) ? A1 : A2;
    const v2f* Wl = ((pass == 0) ? Wp1 : Wp2) + lane;
    const float* Arow = A + (size_t)(mtile * 16 + lrow) * K + 2 * khalf;
    for (int k = 0; k < K; k += 4) {
      v2f a = *(const v2f*)(Arow + k);
      const v2f* Wk = Wl + (size_t)(k >> 2) * (NT * 32);
#pragma unroll
      for (int t = 0; t < NT; ++t) {
        v2f b = Wk[t * 32];             // coalesced b64, immediate offsets
        acc[t] = __builtin_amdgcn_wmma_f32_16x16x4_f32(
            false, a, false, b, (short)0, acc[t], false, false);
      }
    }
  }

#pragma unroll
  for (int t = 0; t < NT; ++t) {
    const int n = t * 16 + lrow;
    if ((NDIM % 16 == 0) || (n < NDIM)) {
      float bb = bias[n];
#pragma unroll
      for (int v = 0; v < 8; ++v) {
        const int m = mtile * 16 + v + 8 * khalf;
        float val = acc[t][v] + bb;
        if (LEAKY) val = (val > 0.f) ? val : LRELU_SLOPE * val;
        size_t idx = (size_t)m * NDIM + n;
        if (ACCUM)
          D[idx] = D[idx] + val;
        else
          D[idx] = val;
      }
    }
  }
}

// ---------------- host launch ----------------

extern "C" void kernel_launch(void* const* d_in, const int* in_sizes, int n_in,
                              void* d_out, int out_size, void* d_ws, size_t ws_size,
                              hipStream_t stream) {
  const int F_IN = 512, H = 128, C = 40;
  const float* x  = (const float*)d_in[0];
  const int* ei   = (const int*)d_in[1];
  const float* ew = (const float*)d_in[2];
  const int N = in_sizes[0] / F_IN;
  const int E = in_sizes[2];
  const int* src = ei;
  const int* dst = ei + E;

  const float* pre_w1 = (const float*)d_in[3];
  const float* pre_b1 = (const float*)d_in[4];
  const float* cw0_1  = (const float*)d_in[5];
  const float* cw1_1  = (const float*)d_in[6];
  const float* cb_1   = (const float*)d_in[7];
  const float* sw_1   = (const float*)d_in[8];
  const float* sb_1   = (const float*)d_in[9];
  const float* lw_1   = (const float*)d_in[10];
  const float* lb_1   = (const float*)d_in[11];
  const float* pre_w2 = (const float*)d_in[12];
  const float* pre_b2 = (const float*)d_in[13];
  const float* cw0_2  = (const float*)d_in[14];
  const float* cw1_2  = (const float*)d_in[15];
  const float* cb_2   = (const float*)d_in[16];
  const float* sw_2   = (const float*)d_in[17];
  const float* sb_2   = (const float*)d_in[18];
  const float* lw_2   = (const float*)d_in[19];
  const float* lb_2   = (const float*)d_in[20];
  const float* cls_w  = (const float*)d_in[21];
  const float* cls_b  = (const float*)d_in[22];

  // workspace layout (floats)
  float* ws     = (float*)d_ws;
  float* dis    = ws;                       // N: deg -> dis
  float* invc   = dis + N;                  // N: cnt -> 1/(cnt+1)
  float* loopw  = invc + N;                 // N
  float* normc  = loopw + N;                // E
  float* wec    = normc + E;                // E
  size_t NH     = (size_t)N * H;
  float* bufA   = wec + E;                  // N*H
  float* bufB   = bufA + NH;                // N*H
  float* bufC   = bufB + NH;                // N*H
  float* bufD   = bufC + NH;                // N*H
  float* logits = bufD + NH;                // N*C
  // packed weights: pack_size(K, NT) = K * NT * 16 floats
  float* p_pre1 = logits + (size_t)N * C;   // 512*8*16 = 65536
  float* p_c01  = p_pre1 + 65536;           // 128*8*16 = 16384 each
  float* p_c11  = p_c01 + 16384;
  float* p_sw1  = p_c11 + 16384;
  float* p_lw1  = p_sw1 + 16384;
  float* p_pre2 = p_lw1 + 16384;
  float* p_c02  = p_pre2 + 16384;
  float* p_c12  = p_c02 + 16384;
  float* p_sw2  = p_c12 + 16384;
  float* p_lw2  = p_sw2 + 16384;
  float* p_cls  = p_lw2 + 16384;            // 128*3*16 = 6144

  const int TB = 256;
  const int gE  = (E + TB - 1) / TB;
  const int gN  = (N + TB - 1) / TB;
  const int gNH = (int)((NH + TB - 1) / TB);
  const int gSC = (E + 7) / 8;              // 8 edges (waves) per block
  const int gCB = (int)(((size_t)N * 32 + TB - 1) / TB);
  const int mtiles = (N + 15) / 16;
  const int gG = (mtiles + 7) / 8;          // 8 waves (M-tiles) per block

  auto pack = [&](const float* W, float* Wp, int K, int NDIM, int NT) {
    int total = (K >> 2) * NT * 32;
    k_pack_w<<<(total + TB - 1) / TB, TB, 0, stream>>>(W, Wp, K, NDIM, NT);
  };

  // ---- weight packing (deterministic, every call) ----
  pack(pre_w1, p_pre1, F_IN, H, 8);
  pack(cw0_1, p_c01, H, H, 8);
  pack(cw1_1, p_c11, H, H, 8);
  pack(sw_1, p_sw1, H, H, 8);
  pack(lw_1, p_lw1, H, H, 8);
  pack(pre_w2, p_pre2, H, H, 8);
  pack(cw0_2, p_c02, H, H, 8);
  pack(cw1_2, p_c12, H, H, 8);
  pack(sw_2, p_sw2, H, H, 8);
  pack(lw_2, p_lw2, H, H, 8);
  pack(cls_w, p_cls, H, C, 3);

  // ---- graph preprocessing ----
  k_fill<<<gN, TB, 0, stream>>>(dis, 0.f, N);
  k_fill<<<gN, TB, 0, stream>>>(invc, 0.f, N);
  k_fill<<<gN, TB, 0, stream>>>(loopw, 1.f, N);
  k_edge_prep<<<gE, TB, 0, stream>>>(src, dst, ew, dis, invc, loopw, E);
  k_node_finalize<<<gN, TB, 0, stream>>>(dis, invc, N);
  k_edge_norm<<<gE, TB, 0, stream>>>(src, dst, ew, dis, normc, wec, E);

  // ---- cell 1 ----
  k_gemm_wmma<8, 128, false, false, false><<<gG, TB, 0, stream>>>(
      x, (const v2f*)p_pre1, nullptr, nullptr, pre_b1, bufA, N, F_IN);
  k_fill<<<gNH, TB, 0, stream>>>(bufB, 0.f, (int)NH);
  k_scatter<<<gSC, TB, 0, stream>>>(src, dst, normc, bufA, bufB, E);
  k_fill<<<gNH, TB, 0, stream>>>(bufC, 0.f, (int)NH);
  k_scatter<<<gSC, TB, 0, stream>>>(src, dst, wec, bufA, bufC, E);
  k_sage_combine<<<gCB, TB, 0, stream>>>(bufC, bufA, loopw, invc, N);
  k_gemm_wmma<8, 128, true, true, false><<<gG, TB, 0, stream>>>(
      bufA, (const v2f*)p_c01, bufB, (const v2f*)p_c11, cb_1, bufD, N, H);
  k_gemm_wmma<8, 128, false, true, true><<<gG, TB, 0, stream>>>(
      bufC, (const v2f*)p_sw1, nullptr, nullptr, sb_1, bufD, N, H);
  k_gemm_wmma<8, 128, false, false, false><<<gG, TB, 0, stream>>>(
      bufD, (const v2f*)p_lw1, nullptr, nullptr, lb_1, bufB, N, H);

  // ---- cell 2 (input = bufB) ----
  k_gemm_wmma<8, 128, false, false, false><<<gG, TB, 0, stream>>>(
      bufB, (const v2f*)p_pre2, nullptr, nullptr, pre_b2, bufA, N, H);
  k_fill<<<gNH, TB, 0, stream>>>(bufC, 0.f, (int)NH);
  k_scatter<<<gSC, TB, 0, stream>>>(src, dst, normc, bufA, bufC, E);
  k_fill<<<gNH, TB, 0, stream>>>(bufD, 0.f, (int)NH);
  k_scatter<<<gSC, TB, 0, stream>>>(src, dst, wec, bufA, bufD, E);
  k_sage_combine<<<gCB, TB, 0, stream>>>(bufD, bufA, loopw, invc, N);
  k_gemm_wmma<8, 128, true, true, false><<<gG, TB, 0, stream>>>(
      bufA, (const v2f*)p_c02, bufC, (const v2f*)p_c12, cb_2, bufB, N, H);
  k_gemm_wmma<8, 128, false, true, true><<<gG, TB, 0, stream>>>(
      bufD, (const v2f*)p_sw2, nullptr, nullptr, sb_2, bufB, N, H);
  k_gemm_wmma<8, 128, false, false, false><<<gG, TB, 0, stream>>>(
      bufB, (const v2f*)p_lw2, nullptr, nullptr, lb_2, bufC, N, H);

  // ---- classifier + log_softmax ----
  k_gemm_wmma<3, 40, false, false, false><<<gG, TB, 0, stream>>>(
      bufC, (const v2f*)p_cls, nullptr, nullptr, cls_b, logits, N, H);
  k_logsoftmax<<<gN, TB, 0, stream>>>(logits, (float*)d_out, N, C);
}